// LocMotionAppearance_57818849738993
// MI455X (gfx1250) — compile-verified
//
#include <hip/hip_runtime.h>

// ---------------- problem constants (from reference) ----------------
#define NB    4
#define LSEG  1200
#define NN    (NB * LSEG)      // 4800 nodes
#define HWDIM 256
#define NPIX  (NB * HWDIM * HWDIM)
#define NE    76800
#define EPSV  1e-5f
#define KC    16               // K-chunk staged in LDS per GEMM iteration

typedef float v8f __attribute__((ext_vector_type(8)));
typedef float v2f __attribute__((ext_vector_type(2)));
typedef __attribute__((address_space(3))) float lds_f;

// ---------------- elementwise / scatter kernels ----------------
__global__ void k_zero(float* p, int n) {
  int i = blockIdx.x * 256 + threadIdx.x;
  if (i < n) p[i] = 0.f;
}

__global__ void k_count_pix(const int* lab, float* cnt) {
  int i = blockIdx.x * 256 + threadIdx.x;
  if (i >= NPIX) return;
  int b = i / (HWDIM * HWDIM);
  atomicAdd(&cnt[b * LSEG + lab[i]], 1.0f);
}

// pool coord(2) + fx(1) + fy(1) + skip0(32) -> 36 features per node (sum)
__global__ void k_pool0(const int* lab, const float* fx, const float* fy,
                        const float* s0, float* acc) {
  int i = blockIdx.x * 256 + threadIdx.x;
  if (i >= NPIX) return;
  int b = i / (HWDIM * HWDIM), rem = i % (HWDIM * HWDIM);
  int r = rem / HWDIM, c = rem % HWDIM;
  float* d = acc + (size_t)(b * LSEG + lab[i]) * 36;
  atomicAdd(d + 0, r * (1.0f / (HWDIM - 1)));   // coord ch0 = row coord
  atomicAdd(d + 1, c * (1.0f / (HWDIM - 1)));   // coord ch1 = col coord
  atomicAdd(d + 2, fx[i]);
  atomicAdd(d + 3, fy[i]);
  const float* sb = s0 + (size_t)b * 32 * HWDIM * HWDIM + rem;
  for (int ch = 0; ch < 32; ++ch)
    atomicAdd(d + 4 + ch, sb[(size_t)ch * HWDIM * HWDIM]);
}

// bilinear-upsample SxS -> 256x256 on the fly, pool into node sums
template <int C, int S>
__global__ void k_pool_up(const int* lab, const float* sk, float* acc) {
  int i = blockIdx.x * 256 + threadIdx.x;
  if (i >= NPIX) return;
  int b = i / (HWDIM * HWDIM), rem = i % (HWDIM * HWDIM);
  int r = rem / HWDIM, c = rem % HWDIM;
  const float scale = (float)(S - 1) / (float)(HWDIM - 1);
  float rf = r * scale, cf = c * scale;
  int r0 = (int)rf, c0 = (int)cf;
  int r1 = min(r0 + 1, S - 1), c1 = min(c0 + 1, S - 1);
  float fr = rf - (float)r0, fc = cf - (float)c0;
  float w00 = (1.f - fr) * (1.f - fc), w01 = (1.f - fr) * fc;
  float w10 = fr * (1.f - fc), w11 = fr * fc;
  const float* base = sk + (size_t)b * C * S * S;
  float* d = acc + (size_t)(b * LSEG + lab[i]) * C;
  for (int ch = 0; ch < C; ++ch) {
    const float* m = base + (size_t)ch * S * S;
    float v = w00 * m[r0 * S + c0] + w01 * m[r0 * S + c1] +
              w10 * m[r1 * S + c0] + w11 * m[r1 * S + c1];
    atomicAdd(d + ch, v);
  }
}

__global__ void k_div_rows(float* buf, const float* cnt, int width, int n) {
  int i = blockIdx.x * 256 + threadIdx.x;
  if (i < n) buf[i] /= fmaxf(cnt[i / width], 1.0f);
}

__global__ void k_edge_cnt(const int* eg, float* cp, float* cn) {
  int e = blockIdx.x * 256 + threadIdx.x;
  if (e >= NE) return;
  int d = eg[NE + e];
  if (eg[2 * NE + e] != -1) atomicAdd(&cp[d], 1.f);
  else                      atomicAdd(&cn[d], 1.f);
}

// segment-sum x[src] (masked by edge sign) into out[dst]; one block per edge
__global__ void k_scatter(float* out, int ldo, int colofs, const float* x,
                          int ldx, int width, const int* eg, int want_pos) {
  int e = blockIdx.x;
  int pos = (eg[2 * NE + e] != -1) ? 1 : 0;
  if (pos != want_pos) return;
  const float* xs = x + (size_t)eg[e] * ldx;
  float* od = out + (size_t)eg[NE + e] * ldo + colofs;
  for (int f = threadIdx.x; f < width; f += blockDim.x)
    atomicAdd(od + f, xs[f]);
}

__global__ void k_bias_init(float* C, int ldc, int colofs, int width,
                            const float* bias, int n) {
  int i = blockIdx.x * 256 + threadIdx.x;
  if (i >= n) return;
  int rr = i / width, f = i % width;
  C[(size_t)rr * ldc + colofs + f] = bias[f];
}

// ---------------- WMMA GEMM: C[NN x m] += A[NN x k] * W[k x m] ----------------
// grid = (NN/64, m/64), block = 128 (4 waves). Each block owns a 64x64 C tile;
// wave w handles rows row0+16w. The KCx64 W-chunk is staged global->LDS with
// CDNA5 async copies (ASYNCcnt), double-buffered so the next chunk streams in
// while the current one feeds V_WMMA_F32_16X16X4_F32 from ds_loads.
__global__ __launch_bounds__(128) void k_gemm_wmma(float* Cm, int ldc,
                                                   const float* A, int lda,
                                                   const float* W, int ldw,
                                                   int kdim) {
  __shared__ float wtile[2][KC * 64];
  const int t = threadIdx.x;
  const int lane = t & 31;
  const int wv = t >> 5;                 // wave id 0..3
  const int lo = lane & 15, hi = lane >> 4;
  const int row0 = blockIdx.x * 64 + wv * 16;
  const int col0 = blockIdx.y * 64;

  v8f c0, c1, c2, c3;
  for (int r = 0; r < 8; ++r) {          // load C (accumulate: bias + prior GEMMs)
    const float* cr = Cm + (size_t)(row0 + r + 8 * hi) * ldc + col0 + lo;
    c0[r] = cr[0]; c1[r] = cr[16]; c2[r] = cr[32]; c3[r] = cr[48];
  }
  const float* arow = A + (size_t)(row0 + lo) * lda;

  const int nchunks = (kdim + KC - 1) / KC;

  // stage chunk kk into wtile[buf]: 256 float4 transfers, 2 per thread.
  auto stage = [&](int buf, int kk) {
    int krem = kdim - kk; if (krem > KC) krem = KC;
    for (int j = 0; j < 2; ++j) {
      int idx = t + 128 * j;             // 0..255
      int rowk = idx >> 4;               // 0..15
      int c4 = (idx & 15) * 4;           // col within tile
      float* lp = &wtile[buf][rowk * 64 + c4];
      if (rowk < krem) {
        const float* g = W + (size_t)(kk + rowk) * ldw + col0 + c4;
        unsigned lds_off = (unsigned)(size_t)(lds_f*)lp;
        unsigned long long ga = (unsigned long long)(size_t)g;
        asm volatile("global_load_async_to_lds_b128 %0, %1, off"
                     :: "v"(lds_off), "v"(ga) : "memory");
      } else {                            // zero-pad ragged K tail
        lp[0] = 0.f; lp[1] = 0.f; lp[2] = 0.f; lp[3] = 0.f;
      }
    }
  };
  auto wait_sync = [&]() {
    asm volatile("s_wait_asynccnt 0x0" ::: "memory");
    __syncthreads();
  };

  stage(0, 0);
  wait_sync();
  for (int ch = 0; ch < nchunks; ++ch) {
    int kk = ch * KC;
    if (ch + 1 < nchunks) stage((ch + 1) & 1, kk + KC);  // prefetch next chunk
    int buf = ch & 1;
    for (int k4 = 0; k4 < KC; k4 += 4) {
      int ka0 = kk + k4 + 2 * hi;
      v2f a;
      a[0] = (ka0 < kdim)     ? arow[ka0]     : 0.f;
      a[1] = (ka0 + 1 < kdim) ? arow[ka0 + 1] : 0.f;
      int bofs = (k4 + 2 * hi) * 64 + lo;
      v2f b;
      b[0] = wtile[buf][bofs];      b[1] = wtile[buf][bofs + 64];
      c0 = __builtin_amdgcn_wmma_f32_16x16x4_f32(false, a, false, b, (short)0, c0, false, false);
      b[0] = wtile[buf][bofs + 16]; b[1] = wtile[buf][bofs + 80];
      c1 = __builtin_amdgcn_wmma_f32_16x16x4_f32(false, a, false, b, (short)0, c1, false, false);
      b[0] = wtile[buf][bofs + 32]; b[1] = wtile[buf][bofs + 96];
      c2 = __builtin_amdgcn_wmma_f32_16x16x4_f32(false, a, false, b, (short)0, c2, false, false);
      b[0] = wtile[buf][bofs + 48]; b[1] = wtile[buf][bofs + 112];
      c3 = __builtin_amdgcn_wmma_f32_16x16x4_f32(false, a, false, b, (short)0, c3, false, false);
    }
    wait_sync();
  }

  for (int r = 0; r < 8; ++r) {
    float* cr = Cm + (size_t)(row0 + r + 8 * hi) * ldc + col0 + lo;
    cr[0] = c0[r]; cr[16] = c1[r]; cr[32] = c2[r]; cr[48] = c3[r];
  }
}

// ---------------- batchnorm ----------------
__global__ void k_bn_stats(const float* x, int ldx, float* mean, float* var) {
  int d = blockIdx.x;
  __shared__ float s1[256], s2[256];
  float a = 0.f, b = 0.f;
  for (int r = threadIdx.x; r < NN; r += 256) {
    float v = x[(size_t)r * ldx + d];
    a += v; b += v * v;
  }
  s1[threadIdx.x] = a; s2[threadIdx.x] = b;
  __syncthreads();
  for (int ofs = 128; ofs > 0; ofs >>= 1) {
    if (threadIdx.x < ofs) {
      s1[threadIdx.x] += s1[threadIdx.x + ofs];
      s2[threadIdx.x] += s2[threadIdx.x + ofs];
    }
    __syncthreads();
  }
  if (threadIdx.x == 0) {
    float m = s1[0] / (float)NN;
    mean[d] = m;
    var[d] = s2[0] / (float)NN - m * m;
  }
}

__global__ void k_bn_relu(float* x, int D, const float* g, const float* bb,
                          const float* mean, const float* var) {
  int i = blockIdx.x * 256 + threadIdx.x;
  if (i >= NN * D) return;
  int d = i % D;
  float v = g[d] * (x[i] - mean[d]) * rsqrtf(var[d] + EPSV) + bb[d];
  x[i] = fmaxf(v, 0.f);
}

__global__ void k_relu(float* x, int n) {
  int i = blockIdx.x * 256 + threadIdx.x;
  if (i < n) x[i] = fmaxf(x[i], 0.f);
}

// ---------------- host orchestration ----------------
extern "C" void kernel_launch(void* const* d_in, const int* in_sizes, int n_in,
                              void* d_out, int out_size, void* d_ws, size_t ws_size,
                              hipStream_t stream) {
  // tensor inputs (setup_inputs dict order)
  const int*   labels = (const int*)d_in[0];
  const float* fx     = (const float*)d_in[1];
  const float* fy     = (const float*)d_in[2];
  const float* skip0  = (const float*)d_in[3];
  const float* skip1  = (const float*)d_in[4];
  const float* skip2  = (const float*)d_in[5];
  const float* skip3  = (const float*)d_in[6];
  const int*   edges  = (const int*)d_in[7];
  // params pytree (sorted dict keys): bn, gcn0..gcn3, lin_pw
  const float* bn_g[4], *bn_b[4];
  for (int l = 0; l < 4; ++l) { bn_g[l] = (const float*)d_in[8 + 2 * l]; bn_b[l] = (const float*)d_in[9 + 2 * l]; }
  // gcn dict sorted: b_neg_r, b_pos_r, w_neg_l, w_neg_r, w_pos_l, w_pos_r
  const float *b_neg[4], *b_pos[4], *w_neg_l[4], *w_neg_r[4], *w_pos_l[4], *w_pos_r[4];
  for (int l = 0; l < 4; ++l) {
    int base = 16 + 6 * l;
    b_neg[l]   = (const float*)d_in[base + 0];
    b_pos[l]   = (const float*)d_in[base + 1];
    w_neg_l[l] = (const float*)d_in[base + 2];
    w_neg_r[l] = (const float*)d_in[base + 3];
    w_pos_l[l] = (const float*)d_in[base + 4];
    w_pos_r[l] = (const float*)d_in[base + 5];
  }
  const float* lin_b = (const float*)d_in[40];
  const float* lin_w = (const float*)d_in[41];

  float* out = (float*)d_out;

  // workspace carve-up (floats)
  float* ws = (float*)d_ws;
  size_t o = 0;
  float* cnt_pix = ws + o; o += NN;
  float* cnt_pos = ws + o; o += NN;
  float* cnt_neg = ws + o; o += NN;
  float* x0  = ws + o; o += (size_t)NN * 36;
  float* sk1 = ws + o; o += (size_t)NN * 64;
  float* sk2 = ws + o; o += (size_t)NN * 128;
  float* sk3 = ws + o; o += (size_t)NN * 256;
  float* mmA = ws + o; o += (size_t)NN * 512;
  float* mmB = ws + o; o += (size_t)NN * 512;
  float* mmC = ws + o; o += (size_t)NN * 512;
  float* mmD = ws + o; o += (size_t)NN * 512;
  float* xA  = ws + o; o += (size_t)NN * 1024;
  float* xB  = ws + o; o += (size_t)NN * 1024;
  float* mean = ws + o; o += 1024;
  float* var  = ws + o; o += 1024;
  (void)o; (void)ws_size; (void)n_in; (void)in_sizes; (void)out_size;

  auto Z = [&](float* p, size_t n) {
    k_zero<<<(unsigned)((n + 255) / 256), 256, 0, stream>>>(p, (int)n);
  };
  auto DIVR = [&](float* buf, const float* cnt, int width) {
    int n = NN * width;
    k_div_rows<<<(n + 255) / 256, 256, 0, stream>>>(buf, cnt, width, n);
  };
  auto GEMM = [&](float* C, int ldc, const float* A, int lda,
                  const float* W, int ldw, int k, int m) {
    dim3 grid(NN / 64, m / 64);
    k_gemm_wmma<<<grid, 128, 0, stream>>>(C, ldc, A, lda, W, ldw, k);
  };
  auto BIAS = [&](float* C, int ldc, int colofs, int width, const float* bias) {
    int n = NN * width;
    k_bias_init<<<(n + 255) / 256, 256, 0, stream>>>(C, ldc, colofs, width, bias, n);
  };

  // ---- superpixel pooling ----
  Z(cnt_pix, NN); Z(cnt_pos, NN); Z(cnt_neg, NN);
  Z(x0, (size_t)NN * 36); Z(sk1, (size_t)NN * 64); Z(sk2, (size_t)NN * 128); Z(sk3, (size_t)NN * 256);
  int pb = NPIX / 256;
  k_count_pix<<<pb, 256, 0, stream>>>(labels, cnt_pix);
  k_pool0<<<pb, 256, 0, stream>>>(labels, fx, fy, skip0, x0);
  k_pool_up<64, 128><<<pb, 256, 0, stream>>>(labels, skip1, sk1);
  k_pool_up<128, 64><<<pb, 256, 0, stream>>>(labels, skip2, sk2);
  k_pool_up<256, 32><<<pb, 256, 0, stream>>>(labels, skip3, sk3);
  DIVR(x0, cnt_pix, 36);
  DIVR(sk1, cnt_pix, 64); DIVR(sk2, cnt_pix, 128); DIVR(sk3, cnt_pix, 256);
  k_edge_cnt<<<(NE + 255) / 256, 256, 0, stream>>>(edges, cnt_pos, cnt_neg);

  // ---- layer 0 (din=36, dout=64 each branch) ----
  Z(mmA, (size_t)NN * 36); Z(mmB, (size_t)NN * 36);
  k_scatter<<<NE, 128, 0, stream>>>(mmA, 36, 0, x0, 36, 36, edges, 1);
  k_scatter<<<NE, 128, 0, stream>>>(mmB, 36, 0, x0, 36, 36, edges, 0);
  DIVR(mmA, cnt_pos, 36); DIVR(mmB, cnt_neg, 36);
  BIAS(xA, 128, 0, 64, b_pos[0]);
  BIAS(xA, 128, 64, 64, b_neg[0]);
  GEMM(xA + 0, 128, mmA, 36, w_pos_l[0], 64, 36, 64);
  GEMM(xA + 0, 128, x0, 36, w_pos_r[0], 64, 36, 64);
  GEMM(xA + 64, 128, mmB, 36, w_neg_l[0], 64, 36, 64);
  GEMM(xA + 64, 128, x0, 36, w_neg_r[0], 64, 36, 64);
  k_bn_stats<<<128, 256, 0, stream>>>(xA, 128, mean, var);
  k_bn_relu<<<(NN * 128 + 255) / 256, 256, 0, stream>>>(xA, 128, bn_g[0], bn_b[0], mean, var);

  // ---- layers 1..3 ----
  float* xin = xA;
  float* xout = xB;
  const float* sks[3] = { sk1, sk2, sk3 };
  for (int li = 1; li <= 3; ++li) {
    int Fh = 32 << li;       // 64, 128, 256 (incoming half width == sk width)
    int F = 2 * Fh;          // 128, 256, 512
    int Din = F;             // incoming x buffer width == 2*Fh == F
    int Dout = 2 * F;
    const float* sk = sks[li - 1];
    // NOTE reference quirk: x_pos = x[:, half:], x_neg = x[:, :half]
    float* xpos = xin + Fh;  // cols [Fh, 2Fh)
    float* xneg = xin;       // cols [0, Fh)

    Z(mmA, (size_t)NN * F); Z(mmB, (size_t)NN * F);
    Z(mmC, (size_t)NN * F); Z(mmD, (size_t)NN * F);
    // mmA = masked_mean(concat(x_pos, sk), pos)
    k_scatter<<<NE, 128, 0, stream>>>(mmA, F, 0, xpos, Din, Fh, edges, 1);
    k_scatter<<<NE, 128, 0, stream>>>(mmA, F, Fh, sk, Fh, Fh, edges, 1);
    // mmB = masked_mean(concat(x_neg, sk), neg)
    k_scatter<<<NE, 128, 0, stream>>>(mmB, F, 0, xneg, Din, Fh, edges, 0);
    k_scatter<<<NE, 128, 0, stream>>>(mmB, F, Fh, sk, Fh, Fh, edges, 0);
    // mmC = masked_mean(concat(x_neg, sk), pos)
    k_scatter<<<NE, 128, 0, stream>>>(mmC, F, 0, xneg, Din, Fh, edges, 1);
    k_scatter<<<NE, 128, 0, stream>>>(mmC, F, Fh, sk, Fh, Fh, edges, 1);
    // mmD = masked_mean(concat(x_pos, sk), neg)
    k_scatter<<<NE, 128, 0, stream>>>(mmD, F, 0, xpos, Din, Fh, edges, 0);
    k_scatter<<<NE, 128, 0, stream>>>(mmD, F, Fh, sk, Fh, Fh, edges, 0);
    DIVR(mmA, cnt_pos, F); DIVR(mmB, cnt_neg, F);
    DIVR(mmC, cnt_pos, F); DIVR(mmD, cnt_neg, F);

    BIAS(xout, Dout, 0, F, b_pos[li]);
    BIAS(xout, Dout, F, F, b_neg[li]);
    // op = [mmA | mmB] @ w_pos_l + concat(x_pos, sk) @ w_pos_r + bias
    GEMM(xout + 0, Dout, mmA, F, w_pos_l[li], F, F, F);
    GEMM(xout + 0, Dout, mmB, F, w_pos_l[li] + (size_t)F * F, F, F, F);
    GEMM(xout + 0, Dout, xpos, Din, w_pos_r[li], F, Fh, F);
    GEMM(xout + 0, Dout, sk, Fh, w_pos_r[li] + (size_t)Fh * F, F, Fh, F);
    // on = [mmC | mmD] @ w_neg_l + concat(x_neg, sk) @ w_neg_r + bias
    GEMM(xout + F, Dout, mmC, F, w_neg_l[li], F, F, F);
    GEMM(xout + F, Dout, mmD, F, w_neg_l[li] + (size_t)F * F, F, F, F);
    GEMM(xout + F, Dout, xneg, Din, w_neg_r[li], F, Fh, F);
    GEMM(xout + F, Dout, sk, Fh, w_neg_r[li] + (size_t)Fh * F, F, Fh, F);

    k_bn_stats<<<Dout, 256, 0, stream>>>(xout, Dout, mean, var);
    k_bn_relu<<<(NN * Dout + 255) / 256, 256, 0, stream>>>(xout, Dout, bn_g[li], bn_b[li], mean, var);

    float* t = xin; xin = xout; xout = t;
  }

  // ---- final: relu(concat(x_pos, x_neg) @ W + b); x_pos = cols[512:1024], x_neg = cols[0:512]
  BIAS(out, 512, 0, 512, lin_b);
  GEMM(out, 512, xin + 512, 1024, lin_w, 512, 512, 512);
  GEMM(out, 512, xin + 0, 1024, lin_w + (size_t)512 * 512, 512, 512, 512);
  k_relu<<<(NN * 512 + 255) / 256, 256, 0, stream>>>(out, NN * 512);
}